// _ProposalLayer_88613765251402
// MI455X (gfx1250) — compile-verified
//
#include <hip/hip_runtime.h>
#include <stdint.h>

#define FEAT_STRIDE 16
#define A_NUM   9
#define H_DIM   32
#define W_DIM   32
#define T_DIM   16
#define HWT     (H_DIM * W_DIM * T_DIM)          // 16384
#define N_PROP  (A_NUM * HWT)                    // 147456
#define NPAD    262144                           // next pow2 >= N_PROP
#define PRE_NMS 6000
#define POST_NMS 300
#define NMS_TH  0.7f
#define WPR     ((PRE_NMS + 31) / 32)            // 188 words per mask row
#define TOPPAD  6016                             // PRE_NMS rounded up to 32
#define BOX_STRIDE 8                             // padded box row (32B, b128-friendly)

#define SORT_CHUNK   8192                        // 64KB LDS of u64 keys
#define SORT_THREADS 512

// ---------------------------------------------------------------------------
// 1) Decode anchors + deltas -> clipped boxes (stride 8), scores, sort keys.
//    key = (~monotone(score) << 32) | index  -> ascending u64 sort ==
//    stable argsort by descending score.
// ---------------------------------------------------------------------------
__global__ void decode_kernel(const float* __restrict__ scores_map,
                              const float* __restrict__ bbox_frame,
                              const float* __restrict__ im_info,
                              const float* __restrict__ anchors,
                              float* __restrict__ boxes,
                              float* __restrict__ scoresFlat,
                              uint64_t* __restrict__ keys)
{
    const int b = blockIdx.y;
    const int n = blockIdx.x * blockDim.x + threadIdx.x;
    if (n >= NPAD) return;
    uint64_t* kb = keys + (size_t)b * NPAD;
    if (n >= N_PROP) { kb[n] = ~0ull; return; }

    const int a = n % A_NUM;
    int r       = n / A_NUM;
    const int t = r % T_DIM;  r /= T_DIM;
    const int w = r % W_DIM;
    const int h = r / W_DIM;

    const size_t sp = ((size_t)h * W_DIM + w) * T_DIM + t;

    const float s = scores_map[(size_t)b * (2 * A_NUM * HWT)
                               + (size_t)(A_NUM + a) * HWT + sp];

    const float* bf = bbox_frame + (size_t)b * (6 * A_NUM * HWT);
    float d[6];
#pragma unroll
    for (int c = 0; c < 6; ++c)
        d[c] = bf[(size_t)(6 * a + c) * HWT + sp];

    const float sx = (float)(w * FEAT_STRIDE);
    const float sy = (float)(h * FEAT_STRIDE);
    const float sz = (float)t;
    const float ax1 = anchors[a * 6 + 0] + sx;
    const float ay1 = anchors[a * 6 + 1] + sy;
    const float az1 = anchors[a * 6 + 2] + sz;
    const float ax2 = anchors[a * 6 + 3] + sx;
    const float ay2 = anchors[a * 6 + 4] + sy;
    const float az2 = anchors[a * 6 + 5] + sz;

    const float aw = ax2 - ax1 + 1.0f;
    const float ah = ay2 - ay1 + 1.0f;
    const float al = az2 - az1 + 1.0f;
    const float acx = ax1 + 0.5f * aw;
    const float acy = ay1 + 0.5f * ah;
    const float acz = az1 + 0.5f * al;

    const float pcx = d[0] * aw + acx;
    const float pcy = d[1] * ah + acy;
    const float pcz = d[2] * al + acz;
    const float pw  = expf(d[3]) * aw;
    const float ph  = expf(d[4]) * ah;
    const float pl  = expf(d[5]) * al;

    const float limx = im_info[1] - 1.0f;
    const float limy = im_info[0] - 1.0f;
    const float limz = im_info[2] - 1.0f;

    const float x1 = fminf(fmaxf(pcx - 0.5f * pw, 0.f), limx);
    const float y1 = fminf(fmaxf(pcy - 0.5f * ph, 0.f), limy);
    const float z1 = fminf(fmaxf(pcz - 0.5f * pl, 0.f), limz);
    const float x2 = fminf(fmaxf(pcx + 0.5f * pw, 0.f), limx);
    const float y2 = fminf(fmaxf(pcy + 0.5f * ph, 0.f), limy);
    const float z2 = fminf(fmaxf(pcz + 0.5f * pl, 0.f), limz);

    float4* bo = (float4*)(boxes + ((size_t)b * N_PROP + n) * BOX_STRIDE);
    bo[0] = make_float4(x1, y1, z1, x2);
    bo[1] = make_float4(y2, z2, 0.f, 0.f);
    scoresFlat[(size_t)b * N_PROP + n] = s;

    const uint32_t u = __float_as_uint(s);
    const uint32_t m = u ^ ((uint32_t)((int32_t)u >> 31) | 0x80000000u);
    kb[n] = ((uint64_t)(~m) << 32) | (uint32_t)n;
}

// ---------------------------------------------------------------------------
// 2) Bitonic sort of NPAD u64 keys per image.  8192-element chunks live in
//    64KB LDS; only j >= 8192 stages touch global (21 launches total).
// ---------------------------------------------------------------------------
__global__ void bitonic_local_init(uint64_t* __restrict__ keys)
{
    __shared__ uint64_t sk[SORT_CHUNK];
    uint64_t* kb = keys + (size_t)blockIdx.y * NPAD
                        + (size_t)blockIdx.x * SORT_CHUNK;
    const unsigned t = threadIdx.x;
    const unsigned gbase = blockIdx.x * SORT_CHUNK;
    for (unsigned e = t; e < SORT_CHUNK; e += SORT_THREADS) sk[e] = kb[e];
    __syncthreads();
    for (unsigned k = 2; k <= SORT_CHUNK; k <<= 1) {
        for (unsigned j = k >> 1; j >= 1; j >>= 1) {
            for (unsigned p = t; p < SORT_CHUNK / 2; p += SORT_THREADS) {
                const unsigned i = ((p & ~(j - 1)) << 1) | (p & (j - 1));
                const unsigned l = i + j;
                const bool up = (((gbase + i) & k) == 0);
                const uint64_t a = sk[i], c = sk[l];
                if (up ? (a > c) : (a < c)) { sk[i] = c; sk[l] = a; }
            }
            __syncthreads();
        }
    }
    for (unsigned e = t; e < SORT_CHUNK; e += SORT_THREADS) kb[e] = sk[e];
}

__global__ void bitonic_local_tail(uint64_t* __restrict__ keys, unsigned k)
{
    __shared__ uint64_t sk[SORT_CHUNK];
    uint64_t* kb = keys + (size_t)blockIdx.y * NPAD
                        + (size_t)blockIdx.x * SORT_CHUNK;
    const unsigned t = threadIdx.x;
    const unsigned gbase = blockIdx.x * SORT_CHUNK;
    for (unsigned e = t; e < SORT_CHUNK; e += SORT_THREADS) sk[e] = kb[e];
    __syncthreads();
    for (unsigned j = SORT_CHUNK / 2; j >= 1; j >>= 1) {
        for (unsigned p = t; p < SORT_CHUNK / 2; p += SORT_THREADS) {
            const unsigned i = ((p & ~(j - 1)) << 1) | (p & (j - 1));
            const unsigned l = i + j;
            const bool up = (((gbase + i) & k) == 0);
            const uint64_t a = sk[i], c = sk[l];
            if (up ? (a > c) : (a < c)) { sk[i] = c; sk[l] = a; }
        }
        __syncthreads();
    }
    for (unsigned e = t; e < SORT_CHUNK; e += SORT_THREADS) kb[e] = sk[e];
}

__global__ void bitonic_global_stage(uint64_t* __restrict__ keys,
                                     unsigned k, unsigned j)
{
    uint64_t* kb = keys + (size_t)blockIdx.y * NPAD;
    const unsigned p = blockIdx.x * blockDim.x + threadIdx.x;  // 0..NPAD/2-1
    const unsigned i = ((p & ~(j - 1)) << 1) | (p & (j - 1));
    const unsigned l = i + j;
    const bool up = ((i & k) == 0);
    const uint64_t a = kb[i], c = kb[l];
    if (up ? (a > c) : (a < c)) { kb[i] = c; kb[l] = a; }
}

// ---------------------------------------------------------------------------
// 3) Gather top PRE_NMS boxes/scores in rank order (pad to TOPPAD with zeros).
// ---------------------------------------------------------------------------
__global__ void gather_top(const uint64_t* __restrict__ keys,
                           const float* __restrict__ boxes,
                           const float* __restrict__ scoresFlat,
                           float* __restrict__ topbox,
                           float* __restrict__ topscore)
{
    const int b = blockIdx.y;
    const int p = blockIdx.x * blockDim.x + threadIdx.x;
    if (p >= TOPPAD) return;
    float4* dst = (float4*)(topbox + ((size_t)b * TOPPAD + p) * BOX_STRIDE);
    if (p < PRE_NMS) {
        const uint32_t idx = (uint32_t)keys[(size_t)b * NPAD + p];
        const float4* src =
            (const float4*)(boxes + ((size_t)b * N_PROP + idx) * BOX_STRIDE);
        dst[0] = src[0];
        dst[1] = src[1];
        topscore[(size_t)b * TOPPAD + p] = scoresFlat[(size_t)b * N_PROP + idx];
    } else {
        dst[0] = make_float4(0.f, 0.f, 0.f, 0.f);
        dst[1] = make_float4(0.f, 0.f, 0.f, 0.f);
        topscore[(size_t)b * TOPPAD + p] = 0.f;
    }
}

// ---------------------------------------------------------------------------
// 4) Pairwise 3D-IoU suppression bitmask.  One wave per 32x32 tile.
//    Column boxes staged into LDS with 2x async b128 DMA per lane (CDNA5).
// ---------------------------------------------------------------------------
__global__ void mask_kernel(const float* __restrict__ topbox,
                            uint32_t* __restrict__ maskbuf)
{
    __shared__ __align__(16) float cb[32 * BOX_STRIDE];
    const int lane = threadIdx.x;
    const int cblk = blockIdx.x;
    const int rblk = blockIdx.y;
    const int b    = blockIdx.z;
    const int r    = rblk * 32 + lane;

    const float* tb = topbox + (size_t)b * TOPPAD * BOX_STRIDE;

    // Async stage of 32 column boxes (32B each) into LDS.  IOFFSET applies to
    // both LDS and global addresses, so one base pair + offsets {0,16} moves
    // the whole padded box.
    {
        const float* gsrc = tb + (size_t)(cblk * 32 + lane) * BOX_STRIDE;
        const uint64_t g  = (uint64_t)(uintptr_t)gsrc;
        const uint32_t l  = (uint32_t)(uintptr_t)(&cb[0]) + (uint32_t)lane * 32u;
        asm volatile("global_load_async_to_lds_b128 %0, %1, off"
                     :: "v"(l), "v"(g) : "memory");
        asm volatile("global_load_async_to_lds_b128 %0, %1, off offset:16"
                     :: "v"(l), "v"(g) : "memory");
    }

    // Overlap: this lane's row box loads while the DMA is in flight.
    const float* rb = tb + (size_t)r * BOX_STRIDE;
    const float4 rlo = *(const float4*)rb;        // x1 y1 z1 x2
    const float2 rhi = *(const float2*)(rb + 4);  // y2 z2
    const float rvol = (rlo.w - rlo.x + 1.f) * (rhi.x - rlo.y + 1.f)
                     * (rhi.y - rlo.z + 1.f);

    asm volatile("s_wait_asynccnt 0" ::: "memory");  // single wave: data ready

    uint32_t word = 0;
#pragma unroll 4
    for (int c = 0; c < 32; ++c) {
        const int col = cblk * 32 + c;
        const float cx1 = cb[c * BOX_STRIDE + 0];
        const float cy1 = cb[c * BOX_STRIDE + 1];
        const float cz1 = cb[c * BOX_STRIDE + 2];
        const float cx2 = cb[c * BOX_STRIDE + 3];
        const float cy2 = cb[c * BOX_STRIDE + 4];
        const float cz2 = cb[c * BOX_STRIDE + 5];
        const float cvol = (cx2 - cx1 + 1.f) * (cy2 - cy1 + 1.f) * (cz2 - cz1 + 1.f);
        const float iw = fmaxf(fminf(rlo.w, cx2) - fmaxf(rlo.x, cx1) + 1.f, 0.f);
        const float ih = fmaxf(fminf(rhi.x, cy2) - fmaxf(rlo.y, cy1) + 1.f, 0.f);
        const float il = fmaxf(fminf(rhi.y, cz2) - fmaxf(rlo.z, cz1) + 1.f, 0.f);
        const float inter = iw * ih * il;
        const float iou = inter / (rvol + cvol - inter);
        const bool sup = (iou > NMS_TH) && (col > r) && (col < PRE_NMS);
        word |= (sup ? 1u : 0u) << c;
    }
    if (r < PRE_NMS)
        maskbuf[((size_t)b * PRE_NMS + r) * (size_t)WPR + cblk] = word;
}

// ---------------------------------------------------------------------------
// 5) Sequential greedy NMS gather pass: one wave32 per image.  Word-level
//    skipping over suppressed candidates + prefetch of upcoming mask rows.
//    Early exit once POST_NMS proposals are kept (output-equivalent).
// ---------------------------------------------------------------------------
__global__ void nms_pass(const uint32_t* __restrict__ maskbuf,
                         const float* __restrict__ topbox,
                         const float* __restrict__ topscore,
                         float* __restrict__ out)
{
    __shared__ uint32_t removed[WPR];
    __shared__ int keepIdx[POST_NMS];
    const int b = blockIdx.x;
    const int lane = threadIdx.x;

    for (int w = lane; w < WPR; w += 32) removed[w] = 0u;
    __syncthreads();

    const uint32_t* mb = maskbuf + (size_t)b * PRE_NMS * WPR;
    int kc = 0;
    int i = 0;
    while (i < PRE_NMS) {
        const int wd = i >> 5;
        const uint32_t avail = ~removed[wd] & (0xFFFFFFFFu << (i & 31));
        if (avail == 0u) { i = (wd + 1) << 5; continue; }
        i = (wd << 5) + (__ffs(avail) - 1);
        if (i >= PRE_NMS) break;               // padding bits of last word
        // keep candidate i
        if (lane == 0) keepIdx[kc] = i;
        ++kc;
        if (kc >= POST_NMS) break;
        // prefetch the next two candidate rows (gfx1250 global_prefetch_b8)
        if (lane < 24 && i + 2 < PRE_NMS) {
            const uint32_t* pf = mb + (size_t)(i + 1 + (lane >= 12 ? 1 : 0)) * WPR
                                    + (size_t)(lane % 12) * 16;
            __builtin_prefetch(pf, 0, 0);
        }
        const uint32_t* mrow = mb + (size_t)i * WPR;
        for (int w = lane; w < WPR; w += 32) removed[w] |= mrow[w];
        __syncthreads();
        ++i;
    }
    __syncthreads();

    for (int k = lane; k < POST_NMS; k += 32) {
        float* row = out + ((size_t)b * POST_NMS + k) * 7;
        if (k < kc) {
            const int p = keepIdx[k];
            row[0] = topscore[(size_t)b * TOPPAD + p];
            const float* tb = topbox + ((size_t)b * TOPPAD + p) * BOX_STRIDE;
#pragma unroll
            for (int c = 0; c < 6; ++c) row[1 + c] = tb[c];
        } else {
#pragma unroll
            for (int c = 0; c < 7; ++c) row[c] = 0.f;
        }
    }
}

// ---------------------------------------------------------------------------
extern "C" void kernel_launch(void* const* d_in, const int* in_sizes, int n_in,
                              void* d_out, int out_size, void* d_ws, size_t ws_size,
                              hipStream_t stream)
{
    const float* scores_map = (const float*)d_in[0];
    const float* bbox_frame = (const float*)d_in[1];
    const float* im_info    = (const float*)d_in[2];
    const float* anchors    = (const float*)d_in[3];
    float* out = (float*)d_out;

    const int B = in_sizes[0] / (2 * A_NUM * HWT);

    uint8_t* ws = (uint8_t*)d_ws;
    size_t off = 0;
    auto alloc = [&](size_t bytes) -> void* {
        void* p = ws + off;
        off = (off + bytes + 255) & ~(size_t)255;
        return p;
    };
    uint64_t* keys    = (uint64_t*)alloc((size_t)B * NPAD * 8);
    float*    boxes   = (float*)   alloc((size_t)B * N_PROP * BOX_STRIDE * 4);
    float*    scoresF = (float*)   alloc((size_t)B * N_PROP * 4);
    float*    topbox  = (float*)   alloc((size_t)B * TOPPAD * BOX_STRIDE * 4);
    float*    topscr  = (float*)   alloc((size_t)B * TOPPAD * 4);
    uint32_t* maskbuf = (uint32_t*)alloc((size_t)B * PRE_NMS * WPR * 4);
    (void)ws_size; (void)n_in; (void)out_size;

    // 1) decode + sort keys
    decode_kernel<<<dim3(NPAD / 256, B), 256, 0, stream>>>(
        scores_map, bbox_frame, im_info, anchors, boxes, scoresF, keys);

    // 2) bitonic sort (ascending u64 == descending score, stable)
    bitonic_local_init<<<dim3(NPAD / SORT_CHUNK, B), SORT_THREADS, 0, stream>>>(keys);
    for (unsigned k = 2 * SORT_CHUNK; k <= NPAD; k <<= 1) {
        for (unsigned j = k >> 1; j >= SORT_CHUNK; j >>= 1)
            bitonic_global_stage<<<dim3(NPAD / 2 / 256, B), 256, 0, stream>>>(keys, k, j);
        bitonic_local_tail<<<dim3(NPAD / SORT_CHUNK, B), SORT_THREADS, 0, stream>>>(keys, k);
    }

    // 3) gather ranked top boxes
    gather_top<<<dim3((TOPPAD + 255) / 256, B), 256, 0, stream>>>(
        keys, boxes, scoresF, topbox, topscr);

    // 4) suppression bitmask (async b128 LDS staging)
    mask_kernel<<<dim3(WPR, WPR, B), 32, 0, stream>>>(topbox, maskbuf);

    // 5) sequential NMS + output
    nms_pass<<<B, 32, 0, stream>>>(maskbuf, topbox, topscr, out);
}